// _BandSplit_70909910057423
// MI455X (gfx1250) — compile-verified
//
#include <hip/hip_runtime.h>
#include <math.h>

#define NB 38
#define B_ 8
#define T_ 2000
#define NCH 25          // stats/t chunks: 2000 = 25 * 80, and 80 = 5 tiles of 16

// ---- subband tables (starts are the prefix sums of widths) ----
__constant__ int kStart[NB] = {
  0,4,8,12,16,20,24,28,32,36,
  40,48,56,64,72,80,88,96,104,112,120,128,
  136,152,168,184,200,216,232,248,264,280,
  296,328,360,392,424,
  456};
__constant__ int kWidth[NB] = {
  4,4,4,4,4,4,4,4,4,4,
  8,8,8,8,8,8,8,8,8,8,8,8,
  16,16,16,16,16,16,16,16,16,16,
  32,32,32,32,32,
  25};
static const int hStart[NB] = {
  0,4,8,12,16,20,24,28,32,36,
  40,48,56,64,72,80,88,96,104,112,120,128,
  136,152,168,184,200,216,232,248,264,280,
  296,328,360,392,424,
  456};

// ---- workspace layout (floats) ----
// WnT (transposed weights, [c][128] per band, base = 512*start[band])
#define WN_OFF   0              // 512*481 = 246272
#define S1_OFF   246272         // NB*128 = 4864
#define S2_OFF   251136         // 4864
#define SUMP_OFF 256000         // NB*B*NCH = 7600
#define SQP_OFF  263600         // 7600
#define MU_OFF   271200         // 304
#define RS_OFF   271504         // 304  -> total 271808 floats (~1.04 MB)

typedef __attribute__((ext_vector_type(2))) float v2f;
typedef __attribute__((ext_vector_type(8))) float v8f;

struct PtrPack { const float* p[NB]; };

// ------------------------------------------------------------------
// Kernel 1: fold norm_w/norm_b into weights.
//  WnT[band][c][o] = fc_w[o,c]*norm_w[c]
//  S2[o] = sum_c Wn[o,c] ; S1[o] = sum_c fc_w[o,c]*norm_b[c] + fc_b[o]
// ------------------------------------------------------------------
__global__ __launch_bounds__(128) void prep_weights(PtrPack fcw, PtrPack nw,
                                                    PtrPack nbv, PtrPack fcb,
                                                    float* __restrict__ ws) {
  const int band = blockIdx.x;
  const int o    = threadIdx.x;      // 0..127
  const int C    = 4 * kWidth[band];
  const float* W  = fcw.p[band];
  const float* g  = nw.p[band];
  const float* be = nbv.p[band];
  float* wnT = ws + WN_OFF + 512 * kStart[band];
  float s1 = 0.f, s2 = 0.f;
  for (int c = 0; c < C; ++c) {
    float wv = W[o * C + c];
    float wn = wv * g[c];
    wnT[c * 128 + o] = wn;
    s2 += wn;
    s1 += wv * be[c];
  }
  ws[S1_OFF + band * 128 + o] = s1 + fcb.p[band][o];
  ws[S2_OFF + band * 128 + o] = s2;
}

// ------------------------------------------------------------------
// Kernel 2: deterministic partial sums/sumsq per (band, b, t-chunk)
// ------------------------------------------------------------------
__global__ __launch_bounds__(256) void stats_partial(const float* __restrict__ inp,
                                                     float* __restrict__ ws) {
  const int chunk = blockIdx.x;   // 0..24
  const int b     = blockIdx.y;   // 0..7
  const int band  = blockIdx.z;   // 0..37
  const int w = kWidth[band], s = kStart[band];
  const int t0 = chunk * 80;
  const int total = 2 * w * 80 * 2;   // m * fo * t * ri
  float sum = 0.f, sq = 0.f;
  for (int j = threadIdx.x; j < total; j += 256) {
    int ri = j & 1;
    int jj = j >> 1;
    int tt = jj % 80;
    int rem = jj / 80;
    int fo = rem % w;
    int m  = rem / w;
    float v = inp[((((size_t)b * 2 + m) * 481 + s + fo) * T_ + t0 + tt) * 2 + ri];
    sum += v; sq += v * v;
  }
  __shared__ float rsum[256], rsq[256];
  rsum[threadIdx.x] = sum; rsq[threadIdx.x] = sq;
  __syncthreads();
  for (int off = 128; off > 0; off >>= 1) {
    if (threadIdx.x < off) {
      rsum[threadIdx.x] += rsum[threadIdx.x + off];
      rsq [threadIdx.x] += rsq [threadIdx.x + off];
    }
    __syncthreads();
  }
  if (threadIdx.x == 0) {
    int idx = (band * B_ + b) * NCH + chunk;
    ws[SUMP_OFF + idx] = rsum[0];
    ws[SQP_OFF  + idx] = rsq[0];
  }
}

// ------------------------------------------------------------------
// Kernel 3: finalize mu / rsqrt(var+eps) per (band, b)
// ------------------------------------------------------------------
__global__ void stats_final(float* __restrict__ ws) {
  const int id = threadIdx.x;           // (band*8 + b)
  if (id >= NB * B_) return;
  const int band = id / B_;
  float sum = 0.f, sq = 0.f;
  for (int c = 0; c < NCH; ++c) {
    sum += ws[SUMP_OFF + id * NCH + c];
    sq  += ws[SQP_OFF  + id * NCH + c];
  }
  const float N = (float)(8000 * kWidth[band]);   // C*T = 4w*2000
  float mu  = sum / N;
  float var = sq / N - mu * mu;
  var = var < 0.f ? 0.f : var;
  ws[MU_OFF + id] = mu;
  ws[RS_OFF + id] = rsqrtf(var + 1e-5f);
}

// ------------------------------------------------------------------
// Kernel 4: fused normalize + 1x1 conv via V_WMMA_F32_16X16X4_F32.
// Block = 8 wave32s; wave w owns output rows [16w,16w+16); block owns
// one (b, band, 80-wide t chunk) = 5 tiles of 16 t.
// y[o,t] = rs * (Wn @ x)[o,t] + (S1[o] - mu*rs*S2[o])
// ------------------------------------------------------------------
template <int C>
__global__ __launch_bounds__(256) void gemm_band(const float* __restrict__ inp,
                                                 float* __restrict__ out,
                                                 const float* __restrict__ ws,
                                                 int bandBase) {
  constexpr int W = C / 4;          // subband width
  constexpr int KSTEPS = C / 4;     // K steps of 4
  const int band = bandBase + blockIdx.y;
  const int b    = blockIdx.z;
  const int s    = kStart[band];
  const int tid  = threadIdx.x;
  const int lane = tid & 31;
  const int wave = tid >> 5;        // 0..7
  const int half = lane >> 4;       // 0/1 (half-wave)
  const int l15  = lane & 15;
  const int obase = wave * 16;

  __shared__ float xs[C * 16];      // x tile, [c][n]

  // A fragments (Wn^T slice) hoisted into registers, reused for 5 t-tiles.
  // A[M][K]: lane = M (l15), v[j] = K = 2*half + j  (ISA 16x4 f32 layout)
  const float* WnT = ws + WN_OFF + 512 * s;   // [C][128]
  v2f afrag[KSTEPS];
#pragma unroll
  for (int ks = 0; ks < KSTEPS; ++ks) {
    const float* ap = WnT + (ks * 4 + 2 * half) * 128 + obase + l15;
    afrag[ks].x = ap[0];
    afrag[ks].y = ap[128];
  }

  const int sb = band * B_ + b;
  const float mu = ws[MU_OFF + sb];
  const float rs = ws[RS_OFF + sb];

  // Per-lane epilogue bias for each accumulator row r: o = obase + r + 8*half
  float bias[8];
#pragma unroll
  for (int r = 0; r < 8; ++r) {
    int o = obase + r + 8 * half;
    bias[r] = ws[S1_OFF + band * 128 + o] - mu * rs * ws[S2_OFF + band * 128 + o];
  }

  for (int tt = 0; tt < 5; ++tt) {
    const int t0 = blockIdx.x * 80 + tt * 16;
    __syncthreads();   // previous tile's LDS reads done
    // Stage x tile: c = (ri*2 + m)*W + fo  (reference channel order)
    for (int j = tid; j < C * 16; j += 256) {
      int c = j >> 4, n = j & 15;
      int fo = c % W;
      int m  = (c / W) & 1;
      int ri = c / (2 * W);
      xs[j] = inp[((((size_t)b * 2 + m) * 481 + s + fo) * T_ + (t0 + n)) * 2 + ri];
    }
    __syncthreads();

    v8f acc = {};
#pragma unroll
    for (int ks = 0; ks < KSTEPS; ++ks) {
      // B[K][N]: lane = N (l15), v[j] = K = 2*half + j
      const float* bp = &xs[(ks * 4 + 2 * half) * 16 + l15];
      v2f bfrag;
      bfrag.x = bp[0];
      bfrag.y = bp[16];
      acc = __builtin_amdgcn_wmma_f32_16x16x4_f32(
          false, afrag[ks], false, bfrag, (short)0, acc, false, false);
    }

    // D[M][N]: v[r] -> o = obase + r + 8*half, t = t0 + l15
    const int t = t0 + l15;
#pragma unroll
    for (int r = 0; r < 8; ++r) {
      int o = obase + r + 8 * half;
      out[(((size_t)b * 128 + o) * NB + band) * T_ + t] = rs * acc[r] + bias[r];
    }
  }
}

// ------------------------------------------------------------------
extern "C" void kernel_launch(void* const* d_in, const int* in_sizes, int n_in,
                              void* d_out, int out_size, void* d_ws, size_t ws_size,
                              hipStream_t stream) {
  const float* inp = (const float*)d_in[0];
  float* out = (float*)d_out;
  float* ws  = (float*)d_ws;

  PtrPack fcw, nw, nbv, fcb;
  if (n_in >= 1 + 4 * NB) {
    // Each list element is its own input: input, norm_w[0..37], norm_b[..], fc_w[..], fc_b[..]
    for (int i = 0; i < NB; ++i) {
      nw.p[i]  = (const float*)d_in[1 + i];
      nbv.p[i] = (const float*)d_in[1 + NB + i];
      fcw.p[i] = (const float*)d_in[1 + 2 * NB + i];
      fcb.p[i] = (const float*)d_in[1 + 3 * NB + i];
    }
  } else {
    // Lists concatenated flat: offsets from prefix sums (C_i = 4*w_i)
    const float* nwc  = (const float*)d_in[1];
    const float* nbc  = (const float*)d_in[2];
    const float* fwc  = (const float*)d_in[3];
    const float* fbc  = (const float*)d_in[4];
    for (int i = 0; i < NB; ++i) {
      nw.p[i]  = nwc + 4 * hStart[i];
      nbv.p[i] = nbc + 4 * hStart[i];
      fcw.p[i] = fwc + 512 * hStart[i];   // 128 * 4 * start
      fcb.p[i] = fbc + 128 * i;
    }
  }

  prep_weights<<<dim3(NB), dim3(128), 0, stream>>>(fcw, nw, nbv, fcb, ws);
  stats_partial<<<dim3(NCH, B_, NB), dim3(256), 0, stream>>>(inp, ws);
  stats_final<<<dim3(1), dim3(320), 0, stream>>>(ws);

  // Bands grouped contiguously by width -> one templated launch per C
  gemm_band<16 ><<<dim3(NCH, 10, B_), dim3(256), 0, stream>>>(inp, out, ws, 0);
  gemm_band<32 ><<<dim3(NCH, 12, B_), dim3(256), 0, stream>>>(inp, out, ws, 10);
  gemm_band<64 ><<<dim3(NCH, 10, B_), dim3(256), 0, stream>>>(inp, out, ws, 22);
  gemm_band<128><<<dim3(NCH,  5, B_), dim3(256), 0, stream>>>(inp, out, ws, 32);
  gemm_band<100><<<dim3(NCH,  1, B_), dim3(256), 0, stream>>>(inp, out, ws, 37);
}